// VQVAE_65094524338950
// MI455X (gfx1250) — compile-verified
//
#include <hip/hip_runtime.h>
#include <hip/hip_bf16.h>

typedef __attribute__((ext_vector_type(2))) float v2f;
typedef __attribute__((ext_vector_type(8))) float v8f;

// ---- CDNA5 f32 WMMA: D(16x16) = A(16x4) * B(4x16) + C ----
// A layout: lane l holds M=l&15, K = (l>>4)*2 + {0,1} in {a.x,a.y}
// B layout: lane l holds N=l&15, K = (l>>4)*2 + {0,1} in {b.x,b.y}
// C/D layout: lane l, elem v -> row M = v + 8*(l>>4), col N = l&15
__device__ __forceinline__ v8f wmma_f32(v2f a, v2f b, v8f c) {
    return __builtin_amdgcn_wmma_f32_16x16x4_f32(false, a, false, b, (short)0, c,
                                                 false, false);
}

// ---------------- sizes ----------------
#define NIMG 64
#define H0 253
#define C1 128
#define H1 127
#define C2 64
#define H2 64
#define K_ENC2 1152   // 128*9
#define K_DEC1 576    // 64*9
#define NCODE 512
#define EMB 64
#define NROWS (64LL*64*64)            // 262144 VQ rows
#define ZE_ELEMS (64LL*64*64*64)      // 16777216
#define H1_ELEMS (64LL*128*127*127)   // 132128768
#define XR_ELEMS (64LL*253*253)       // 4096576
#define PITCH 148                     // 144 + 4 pad -> conflict-free column reads

// ================= enc1: direct conv 1->128 s2 + ReLU =================
__global__ void enc1_conv(const float* __restrict__ x, const float* __restrict__ w,
                          const float* __restrict__ b, float* __restrict__ h1) {
    long long i = (long long)blockIdx.x * 256 + threadIdx.x;
    if (i >= H1_ELEMS) return;
    int xw = (int)(i % H1); long long t = i / H1;
    int yh = (int)(t % H1); t /= H1;
    int co = (int)(t % C1); int n = (int)(t / C1);
    float acc = b[co];
    const float* wr = w + co * 9;
    const float* xp = x + (long long)n * H0 * H0;
    int iy0 = 2 * yh - 1, ix0 = 2 * xw - 1;
#pragma unroll
    for (int ty = 0; ty < 3; ty++) {
        int iy = iy0 + ty; if (iy < 0 || iy >= H0) continue;
#pragma unroll
        for (int tx = 0; tx < 3; tx++) {
            int ix = ix0 + tx; if (ix < 0 || ix >= H0) continue;
            acc = fmaf(wr[ty * 3 + tx], xp[(long long)iy * H0 + ix], acc);
        }
    }
    h1[i] = fmaxf(acc, 0.f);
}

// ======== enc2: implicit-GEMM WMMA, M=16 pixels, N=64 (4 tiles), K=1152 ========
__global__ void __launch_bounds__(32)
enc2_wmma(const float* __restrict__ h1, const float* __restrict__ w,
          const float* __restrict__ bias, float* __restrict__ ze) {
    __shared__ float A[16 * PITCH];
    const int lane = threadIdx.x, half = lane >> 4, lm = lane & 15;
    const long long pbase = (long long)blockIdx.x * 16;  // pixel id: n*4096 + y*64 + x
    v8f acc[4] = {};
    for (int ch = 0; ch < 8; ch++) {           // 8 chunks of 144 K-values (16 ci each)
        for (int v = lane; v < 16 * 144; v += 32) {
            int pix = v / 144, kk = v % 144;
            int k = ch * 144 + kk;
            int ci = k / 9, tap = k % 9, ty = tap / 3, tx = tap % 3;
            long long pid = pbase + pix;
            int n = (int)(pid >> 12), rem = (int)(pid & 4095);
            int y = rem >> 6, xx = rem & 63;
            int iy = 2 * y - 1 + ty, ix = 2 * xx - 1 + tx;
            float val = 0.f;
            if (iy >= 0 && iy < H1 && ix >= 0 && ix < H1)
                val = h1[(((long long)n * C1 + ci) * H1 + iy) * H1 + ix];
            A[pix * PITCH + kk] = val;
        }
        __syncthreads();
        for (int kk = 0; kk < 144; kk += 4) {
            int Kb = kk + half * 2;
            v2f a; a.x = A[lm * PITCH + Kb]; a.y = A[lm * PITCH + Kb + 1];
            int kg = ch * 144 + Kb;
#pragma unroll
            for (int nt = 0; nt < 4; nt++) {
                const float* wp = w + (size_t)(nt * 16 + lm) * K_ENC2 + kg;
                v2f b; b.x = wp[0]; b.y = wp[1];
                acc[nt] = wmma_f32(a, b, acc[nt]);
            }
        }
        __syncthreads();
    }
#pragma unroll
    for (int v = 0; v < 8; v++) {
        int M = v + half * 8;
        long long pid = pbase + M;
        int n = (int)(pid >> 12), rem = (int)(pid & 4095);
        int y = rem >> 6, xx = rem & 63;
#pragma unroll
        for (int nt = 0; nt < 4; nt++) {
            int co = nt * 16 + lm;
            ze[(((long long)n * C2 + co) * H2 + y) * H2 + xx] = acc[nt][v] + bias[co];
        }
    }
}

// ================= 0.5*||e_c||^2 for each code =================
__global__ void cnorm_k(const float* __restrict__ cb, float* __restrict__ cn) {
    int c = blockIdx.x * 64 + threadIdx.x;
    if (c >= NCODE) return;
    float s = 0.f;
    for (int d = 0; d < EMB; d++) { float e = cb[c * EMB + d]; s += e * e; }
    cn[c] = 0.5f * s;
}

// ======== VQ argmin: scores via WMMA GEMM (16 rows x 512 codes, K=64) ========
#define VQP 68
__global__ void __launch_bounds__(32)
vq_argmin(const float* __restrict__ ze, const float* __restrict__ cb,
          const float* __restrict__ cn, int* __restrict__ idx) {
    __shared__ float A[16 * VQP];
    const int lane = threadIdx.x, half = lane >> 4, lm = lane & 15;
    const long long rbase = (long long)blockIdx.x * 16;
    for (int v = lane; v < 16 * EMB; v += 32) {
        int r = v >> 6, d = v & 63;
        A[r * VQP + d] = ze[(rbase + r) * EMB + d];
    }
    __syncthreads();
    float best[8]; int bidx[8];
#pragma unroll
    for (int v = 0; v < 8; v++) { best[v] = 3.4e38f; bidx[v] = 0; }
    for (int nt = 0; nt < NCODE / 16; nt++) {
        v8f acc = {};
        int c = nt * 16 + lm;
        for (int kk = 0; kk < EMB; kk += 4) {
            int Kb = kk + half * 2;
            v2f a; a.x = A[lm * VQP + Kb]; a.y = A[lm * VQP + Kb + 1];
            v2f b; b.x = cb[c * EMB + Kb]; b.y = cb[c * EMB + Kb + 1];
            acc = wmma_f32(a, b, acc);
        }
        float cno = cn[c];
#pragma unroll
        for (int v = 0; v < 8; v++) {
            float score = cno - acc[v];            // argmin of 0.5|e|^2 - z.e
            if (score < best[v]) { best[v] = score; bidx[v] = c; }
        }
    }
#pragma unroll
    for (int v = 0; v < 8; v++) {                  // min+argmin across 16 lanes
        float bv = best[v]; int bi = bidx[v];
        for (int off = 8; off >= 1; off >>= 1) {
            float ov = __shfl_xor(bv, off, 16);
            int   oi = __shfl_xor(bi, off, 16);
            if (ov < bv) { bv = ov; bi = oi; }
        }
        if (lm == 0) idx[rbase + v + half * 8] = bi;
    }
}

// ====== gather z_q = codebook[idx], accumulate sum((q - z_e)^2) ======
__global__ void vq_gather(const float* __restrict__ ze, const float* __restrict__ cb,
                          const int* __restrict__ idx, float* __restrict__ zq,
                          float* __restrict__ accum) {
    long long i = (long long)blockIdx.x * 256 + threadIdx.x;
    float d2 = 0.f;
    if (i < ZE_ELEMS) {
        long long r = i >> 6; int d = (int)(i & 63);
        float q = cb[(size_t)idx[r] * EMB + d];
        zq[i] = q;
        float df = q - ze[i];
        d2 = df * df;
    }
    __shared__ float red[256];
    red[threadIdx.x] = d2; __syncthreads();
    for (int s = 128; s > 0; s >>= 1) {
        if (threadIdx.x < s) red[threadIdx.x] += red[threadIdx.x + s];
        __syncthreads();
    }
    if (threadIdx.x == 0) atomicAdd(&accum[0], red[0]);
}

// ==== dec1: ConvT 64->128 as implicit-GEMM WMMA over dilated input, K=576 ====
__global__ void __launch_bounds__(32)
dec1_wmma(const float* __restrict__ zq, const float* __restrict__ w,
          const float* __restrict__ bias, float* __restrict__ h3) {
    __shared__ float A[16 * PITCH];
    const int lane = threadIdx.x, half = lane >> 4, lm = lane & 15;
    const long long pbase = (long long)blockIdx.x * 16;  // pid over 64*127*127
    v8f acc[8] = {};
    for (int ch = 0; ch < 4; ch++) {           // 4 chunks of 144 (16 ci each)
        for (int v = lane; v < 16 * 144; v += 32) {
            int pix = v / 144, kk = v % 144;
            int k = ch * 144 + kk;
            int ci = k / 9, tap = k % 9, ty = tap / 3, tx = tap % 3;
            long long pid = pbase + pix;
            int n = (int)(pid / 16129), rem = (int)(pid % 16129);
            int oy = rem / 127, ox = rem % 127;
            int yy = oy - 1 + ty, xx = ox - 1 + tx;
            float val = 0.f;
            if (yy >= 0 && xx >= 0 && !(yy & 1) && !(xx & 1)) {
                int y = yy >> 1, x = xx >> 1;
                if (y < H2 && x < H2)
                    val = zq[(((long long)n * C2 + ci) * H2 + y) * H2 + x];
            }
            A[pix * PITCH + kk] = val;
        }
        __syncthreads();
        for (int kk = 0; kk < 144; kk += 4) {
            int Kb = kk + half * 2;
            v2f a; a.x = A[lm * PITCH + Kb]; a.y = A[lm * PITCH + Kb + 1];
            int k0 = ch * 144 + Kb, k1 = k0 + 1;
            int ci0 = k0 / 9, t0 = k0 % 9, ci1 = k1 / 9, t1 = k1 % 9;
            int w0 = (2 - t0 / 3) * 3 + (2 - t0 % 3);   // spatial flip
            int w1 = (2 - t1 / 3) * 3 + (2 - t1 % 3);
#pragma unroll
            for (int nt = 0; nt < 8; nt++) {
                int co = nt * 16 + lm;
                v2f b;
                b.x = w[((size_t)ci0 * C1 + co) * 9 + w0];   // dec1_w[ci][co][2-ty][2-tx]
                b.y = w[((size_t)ci1 * C1 + co) * 9 + w1];
                acc[nt] = wmma_f32(a, b, acc[nt]);
            }
        }
        __syncthreads();
    }
#pragma unroll
    for (int v = 0; v < 8; v++) {
        long long pid = pbase + v + half * 8;
        int n = (int)(pid / 16129), rem = (int)(pid % 16129);
        int oy = rem / 127, ox = rem % 127;
#pragma unroll
        for (int nt = 0; nt < 8; nt++) {
            int co = nt * 16 + lm;
            float r = acc[nt][v] + bias[co];
            h3[(((long long)n * C1 + co) * H1 + oy) * H1 + ox] = fmaxf(r, 0.f);
        }
    }
}

// ==== dec2: ConvT 128->1 + sigmoid, plus recon-loss accumulation ====
__global__ void dec2_sigmoid(const float* __restrict__ h3, const float* __restrict__ w,
                             const float* __restrict__ bias, const float* __restrict__ x,
                             float* __restrict__ out, float* __restrict__ accum) {
    long long i = (long long)blockIdx.x * 256 + threadIdx.x;
    float d2 = 0.f;
    if (i < XR_ELEMS) {
        int ox = (int)(i % H0); long long t = i / H0;
        int oy = (int)(t % H0); int n = (int)(t / H0);
        float acc = bias[0];
#pragma unroll
        for (int ty = 0; ty < 3; ty++) {
            int yy = oy - 1 + ty; if (yy < 0 || (yy & 1)) continue;
            int y = yy >> 1; if (y >= H1) continue;
#pragma unroll
            for (int tx = 0; tx < 3; tx++) {
                int xx = ox - 1 + tx; if (xx < 0 || (xx & 1)) continue;
                int xq = xx >> 1; if (xq >= H1) continue;
                int widx = (2 - ty) * 3 + (2 - tx);
                const float* hp = h3 + ((long long)n * C1 * H1 + y) * H1 + xq;
                for (int ci = 0; ci < C1; ci++)
                    acc = fmaf(w[ci * 9 + widx], hp[(long long)ci * H1 * H1], acc);
            }
        }
        float s = 1.f / (1.f + expf(-acc));
        out[i] = s;
        float df = s - x[i];
        d2 = df * df;
    }
    __shared__ float red[256];
    red[threadIdx.x] = d2; __syncthreads();
    for (int s = 128; s > 0; s >>= 1) {
        if (threadIdx.x < s) red[threadIdx.x] += red[threadIdx.x + s];
        __syncthreads();
    }
    if (threadIdx.x == 0) atomicAdd(&accum[1], red[0]);
}

__global__ void init_accum(float* a) { if (threadIdx.x < 2) a[threadIdx.x] = 0.f; }

__global__ void finalize_loss(const float* __restrict__ accum, float* __restrict__ out) {
    // loss = recon_mean + (1 + 0.25) * mean((q - z_e)^2)   [stop_gradient is identity fwd]
    out[XR_ELEMS] = accum[1] / (float)XR_ELEMS + 1.25f * (accum[0] / (float)ZE_ELEMS);
}

extern "C" void kernel_launch(void* const* d_in, const int* in_sizes, int n_in,
                              void* d_out, int out_size, void* d_ws, size_t ws_size,
                              hipStream_t stream) {
    const float* x       = (const float*)d_in[0];
    const float* enc1_w  = (const float*)d_in[1];
    const float* enc1_b  = (const float*)d_in[2];
    const float* enc2_w  = (const float*)d_in[3];
    const float* enc2_b  = (const float*)d_in[4];
    const float* cb      = (const float*)d_in[5];
    const float* dec1_w  = (const float*)d_in[6];
    const float* dec1_b  = (const float*)d_in[7];
    const float* dec2_w  = (const float*)d_in[8];
    const float* dec2_b  = (const float*)d_in[9];
    float* out = (float*)d_out;

    // workspace layout
    char* ws = (char*)d_ws;
    size_t off = 0;
    float* accum = (float*)(ws + off); off += 256;
    float* h1    = (float*)(ws + off); off += (size_t)H1_ELEMS * 4;  // aliased by h3
    float* ze    = (float*)(ws + off); off += (size_t)ZE_ELEMS * 4;
    float* zq    = (float*)(ws + off); off += (size_t)ZE_ELEMS * 4;
    int*   idx   = (int*)  (ws + off); off += (size_t)NROWS * 4;
    float* cn    = (float*)(ws + off); off += (size_t)NCODE * 4;
    float* h3 = h1;  // h1 dead after enc2

    init_accum<<<1, 64, 0, stream>>>(accum);

    {   // enc1
        long long total = H1_ELEMS;
        int blocks = (int)((total + 255) / 256);
        enc1_conv<<<blocks, 256, 0, stream>>>(x, enc1_w, enc1_b, h1);
    }
    enc2_wmma<<<(int)(NROWS / 16), 32, 0, stream>>>(h1, enc2_w, enc2_b, ze);
    cnorm_k<<<NCODE / 64, 64, 0, stream>>>(cb, cn);
    vq_argmin<<<(int)(NROWS / 16), 32, 0, stream>>>(ze, cb, cn, idx);
    vq_gather<<<(int)(ZE_ELEMS / 256), 256, 0, stream>>>(ze, cb, idx, zq, accum);
    dec1_wmma<<<(int)(64LL * 16129 / 16), 32, 0, stream>>>(zq, dec1_w, dec1_b, h3);
    {   // dec2 + recon loss
        int blocks = (int)((XR_ELEMS + 255) / 256);
        dec2_sigmoid<<<blocks, 256, 0, stream>>>(h3, dec2_w, dec2_b, x, out, accum);
    }
    finalize_loss<<<1, 1, 0, stream>>>(accum, out);
}